// HuberLoss_9740985827461
// MI455X (gfx1250) — compile-verified
//
#include <hip/hip_runtime.h>
#include <hip/hip_bf16.h>

// Huber loss mean over 67.1M f32 elements: pure HBM-bandwidth-bound reduction.
// Phase 1: streaming partial sums (b128 non-temporal loads, wave32 reductions).
// Phase 2: single-wave epilogue using V_WMMA_F32_16X16X4_F32 (A=partials,
//          B=ones) as the cross-lane summation engine, f32 throughout.

typedef float f4  __attribute__((ext_vector_type(4)));
typedef float v2f __attribute__((ext_vector_type(2)));
typedef float v8f __attribute__((ext_vector_type(8)));

#define BLOCK 256
#define NWAVES (BLOCK / 32)

__global__ void huber_partial_kernel(const float* __restrict__ x,
                                     const float* __restrict__ y,
                                     const float* __restrict__ delta_ptr,
                                     float* __restrict__ partials,
                                     long long n) {
    const float delta = delta_ptr[0];
    const float half_delta = 0.5f * delta;

    const long long n4     = n >> 2;
    const long long tid    = (long long)blockIdx.x * blockDim.x + threadIdx.x;
    const long long stride = (long long)gridDim.x * blockDim.x;

    const f4* __restrict__ x4 = (const f4*)x;
    const f4* __restrict__ y4 = (const f4*)y;

    float acc = 0.0f;

    // Main streaming loop: two global_load_b128 per iteration, NT hint
    // (read-once stream; keep the 192MB L2 clean).
    for (long long i = tid; i < n4; i += stride) {
        f4 xv = __builtin_nontemporal_load(&x4[i]);
        f4 yv = __builtin_nontemporal_load(&y4[i]);
#pragma unroll
        for (int c = 0; c < 4; ++c) {
            float a   = fabsf(xv[c] - yv[c]);
            float sq  = 0.5f * a * a;
            float lin = delta * (a - half_delta);
            acc += (a <= delta) ? sq : lin;
        }
    }
    // Scalar tail (n not divisible by 4) — generality; no-op for this shape.
    for (long long i = (n4 << 2) + tid; i < n; i += stride) {
        float a = fabsf(x[i] - y[i]);
        acc += (a <= delta) ? 0.5f * a * a : delta * (a - half_delta);
    }

    // Wave32 reduction.
#pragma unroll
    for (int off = 16; off > 0; off >>= 1)
        acc += __shfl_xor(acc, off, 32);

    // Cross-wave reduction via LDS.
    __shared__ float smem[NWAVES];
    const int lane = threadIdx.x & 31;
    const int wave = threadIdx.x >> 5;
    if (lane == 0) smem[wave] = acc;
    __syncthreads();
    if (wave == 0) {
        float v = (lane < NWAVES) ? smem[lane] : 0.0f;
#pragma unroll
        for (int off = 16; off > 0; off >>= 1)
            v += __shfl_xor(v, off, 32);
        if (lane == 0) partials[blockIdx.x] = v;
    }
}

// One wave (32 threads, EXEC all-ones as WMMA requires).
// Cross-lane sum done with V_WMMA_F32_16X16X4_F32:
//   A (16x4): lane m holds {s_m, 0}; lane m+16 holds A[m][2..3] = {s_{m+16}, 0}
//   B (4x16): all ones
//   D[m][n]  = s_m + s_{m+16}  (independent of n)
// Lane n in 0..15:  c[j] = D[j][n]   -> t  = sum_{m=0..7} (s_m + s_{m+16})
// Lane n in 16..31: c[j] = D[j+8][n] -> t' = sum_{m=8..15}(s_m + s_{m+16})
// total = t + shfl_xor(t, 16).
__global__ void huber_final_kernel(const float* __restrict__ partials,
                                   int nparts,
                                   float* __restrict__ out,
                                   float inv_n) {
    const int lane = threadIdx.x;  // 0..31
    float s = 0.0f;
    for (int i = lane; i < nparts; i += 32) s += partials[i];

    v2f a; a[0] = s;    a[1] = 0.0f;
    v2f b; b[0] = 1.0f; b[1] = 1.0f;
    v8f c = {};
    c = __builtin_amdgcn_wmma_f32_16x16x4_f32(
        /*neg_a=*/false, a, /*neg_b=*/false, b,
        /*c_mod=*/(short)0, c, /*reuse_a=*/false, /*reuse_b=*/false);

    float t = c[0] + c[1] + c[2] + c[3] + c[4] + c[5] + c[6] + c[7];
    t += __shfl_xor(t, 16, 32);

    if (lane == 0) out[0] = t * inv_n;
}

extern "C" void kernel_launch(void* const* d_in, const int* in_sizes, int n_in,
                              void* d_out, int out_size, void* d_ws, size_t ws_size,
                              hipStream_t stream) {
    (void)n_in; (void)out_size;

    const float* x     = (const float*)d_in[0];
    const float* y     = (const float*)d_in[1];
    const float* delta = (const float*)d_in[2];
    float*       out   = (float*)d_out;
    float*       part  = (float*)d_ws;

    const long long n = (long long)in_sizes[0];

    // 2048 blocks x 256 threads: 16K wave32s, 32 float4-pairs per thread.
    int nblocks = 2048;
    size_t max_parts = ws_size / sizeof(float);
    if ((size_t)nblocks > max_parts) nblocks = (int)max_parts;
    if (nblocks < 1) nblocks = 1;

    huber_partial_kernel<<<nblocks, BLOCK, 0, stream>>>(x, y, delta, part, n);
    huber_final_kernel<<<1, 32, 0, stream>>>(part, nblocks, out,
                                             1.0f / (float)n);
}